// RansacRouting_3118146257452
// MI455X (gfx1250) — compile-verified
//
#include <hip/hip_runtime.h>

// RANSAC capsule routing, fused single kernel for MI455X (gfx1250, wave32).
// v[b,o,:] == Mu[b,o,:,argmin_h losses] (exact algebraic reduction of reference).
// One block per (b,o): TDM (tensor_load_to_lds) stages u[b,:,o,:] into LDS with
// hardware row padding while all waves scatter the hypothesis bitmask; both
// inner GEMMs run on V_WMMA_F32_16X16X4_F32.

typedef __attribute__((ext_vector_type(2))) float    v2f;
typedef __attribute__((ext_vector_type(8))) float    v8f;
typedef __attribute__((ext_vector_type(4))) unsigned v4u;
typedef __attribute__((ext_vector_type(8))) int      v8i;
typedef __attribute__((ext_vector_type(4))) int      v4i;

constexpr int kB = 32;
constexpr int kI = 1152;
constexpr int kO = 10;
constexpr int kD = 16;
constexpr int kH = 10;     // hypotheses (padded to 16 in WMMA tiles)
constexpr int kS = 922;    // subset size
constexpr int kThreads = 256;  // 8 wave32
constexpr int kWaves = 8;
constexpr int kDStride = 20;        // LDS row stride (floats): 16 data + 4 pad (TDM pad_amount)
constexpr int kKSteps = kI / 4;     // 288 wmma k-steps for num GEMM
constexpr int kTiles  = kI / 16;    // 72 i-tiles for loss GEMM

__global__ __launch_bounds__(kThreads)
void ransac_routing_kernel(const float* __restrict__ u_predict,
                           const int*  __restrict__ sample_idx,
                           float*      __restrict__ out)
{
  __shared__ __align__(16) float sU[kI * kDStride];   // u[b,:,o,:], rows padded by TDM
  __shared__ float    sVn[kI];                        // ||u_i||
  __shared__ unsigned sMask[kI];                      // bit h set <=> r[i,h]==1
  __shared__ float    sPart[kWaves][16][16];          // per-wave num partials
  __shared__ float    sMu[16][16];                    // Mu[h][d]
  __shared__ float    sDen[16];
  __shared__ float    sMuN2[16];                      // ||Mu_h||^2
  __shared__ float    sLoss[16];
  __shared__ int      sBest;

  const int bo   = blockIdx.x;          // 0..319
  const int b    = bo / kO;
  const int o    = bo - b * kO;
  const int tid  = threadIdx.x;
  const int lane = tid & 31;
  const int wave = tid >> 5;
  const int l15  = lane & 15;
  const int hi2  = (lane >> 4) << 1;    // 0 or 2 (wmma K half-select)
  const int hi8  = (lane >> 4) << 3;    // 0 or 8 (wmma M half-select for C)

  // ---------- Phase 1a: wave 0 kicks off the TDM DMA of u[b,:,o,:] -> sU
  // 2D tile: 1152 rows x 16 f32, row stride 160 f32; LDS pad 4 DWORDs per 16
  // DWORDs stored -> 20-float padded rows (bank-conflict-free for WMMA fetches).
  const float* ubase = u_predict + ((size_t)b * kI * kO + (size_t)o) * kD;
  if (wave == 0) {
    const unsigned long long gaddr = (unsigned long long)(uintptr_t)ubase;
    const unsigned lds_off = (unsigned)(uintptr_t)(&sU[0]);  // aperture low 32 = LDS byte offset

    v4u g0;
    g0.x = 1u;                                   // count=1 valid descriptor
    g0.y = lds_off;                              // lds_addr [63:32]
    g0.z = (unsigned)(gaddr & 0xFFFFFFFFull);    // global_addr[31:0]
    g0.w = (unsigned)((gaddr >> 32) & 0x01FFFFFFull) | (2u << 30);  // addr[56:32] | type=2

    v8i g1;
    g1[0] = (int)((2u << 16)      // data_size = 4B
                | (1u << 20)      // pad_enable
                | (3u << 22)      // pad_interval: 16 DWORDs between pads
                | (3u << 25));    // pad_amount: 4 DWORDs
    g1[1] = (int)(16u << 16);     // tensor_dim0[15:0]=16 at bits 63:48
    g1[2] = (int)(1152u << 16);   // tensor_dim1[15:0]=1152 at bits 95:80
    g1[3] = (int)(16u << 16);     // tile_dim0=16 at bits 127:112
    g1[4] = (int)1152u;           // tile_dim1=1152 at bits 143:128
    g1[5] = (int)160u;            // tensor_dim0_stride=160 at bits 191:160
    g1[6] = 0;                    // stride hi / tensor_dim1_stride lo
    g1[7] = 0;

    v4i gz4 = {0, 0, 0, 0};                      // groups 2/3 unused for 2D tensors
    v8i gz8 = {0, 0, 0, 0, 0, 0, 0, 0};          // trailing group (clang-23 6-arg form)
    __builtin_amdgcn_tensor_load_to_lds(g0, g1, gz4, gz4, gz8, 0);
  }

  // ---------- Phase 1b: (overlapped with TDM) zero mask, init accumulators
  for (int t = tid; t < kI; t += kThreads) sMask[t] = 0u;
  if (tid < 16) { sDen[tid] = 0.f; sLoss[tid] = 0.f; }
  __syncthreads();

  // ---------- Phase 2a: hypothesis bitmask scatter (global gathers overlap the DMA)
  const int* sidx = sample_idx + (size_t)b * kS * kO * kH + (size_t)o * kH;
  for (int t = tid; t < kS * kH; t += kThreads) {
    const int s = t / kH, h = t - s * kH;
    const int i = sidx[(size_t)s * (kO * kH) + h];
    atomicOr(&sMask[i], 1u << h);
  }

  // ---------- Phase 2b: publish the TDM tile, then Vnorm (zero-safe)
  if (wave == 0) __builtin_amdgcn_s_wait_tensorcnt(0);
  __syncthreads();

  for (int i = tid; i < kI; i += kThreads) {
    float s = 0.f;
#pragma unroll
    for (int d = 0; d < kD; ++d) { const float x = sU[i * kDStride + d]; s += x * x; }
    sVn[i] = (s == 0.f) ? 0.f : __builtin_sqrtf(s);
  }
  __syncthreads();

  // ---------- Phase 3: num[h][d] = sum_i w[h,i]*u[i,d]  (WMMA, K split over waves)
  // w[h,i] = r[i,h]*Vnorm[i], generated branchlessly (cvt+mul, no EXEC dance).
  {
    v8f c = {0.f, 0.f, 0.f, 0.f, 0.f, 0.f, 0.f, 0.f};
    const int kPerWave = kKSteps / kWaves;  // 36
    for (int ks = wave * kPerWave; ks < (wave + 1) * kPerWave; ++ks) {
      const int k0 = (ks << 2) + hi2;       // this lane's two K indices: k0, k0+1
      const unsigned m0 = sMask[k0], m1 = sMask[k0 + 1];
      const float vn0 = sVn[k0], vn1 = sVn[k0 + 1];
      v2f a, bm;
      a.x = vn0 * (float)((m0 >> l15) & 1u);   // A[M=h=l15][K=i]
      a.y = vn1 * (float)((m1 >> l15) & 1u);
      bm.x = sU[k0 * kDStride + l15];          // B[K=i][N=d=l15]
      bm.y = sU[(k0 + 1) * kDStride + l15];
      c = __builtin_amdgcn_wmma_f32_16x16x4_f32(false, a, false, bm,
                                                (short)0, c, false, false);
    }
#pragma unroll
    for (int v = 0; v < 8; ++v)
      sPart[wave][v + hi8][l15] = c[v];        // C[M=v+hi8][N=l15]
  }
  // den[h] = sum_i w[h,i]  (16 chunks of 72 per h, branchless, LDS float atomics)
  {
    const int h = tid >> 4, ch = tid & 15;
    if (h < kH) {
      float s = 0.f;
      const int i0 = ch * (kI / 16);
      for (int i = i0; i < i0 + kI / 16; ++i)
        s += sVn[i] * (float)((sMask[i] >> h) & 1u);
      atomicAdd(&sDen[h], s);
    }
  }
  __syncthreads();

  // ---------- Phase 4: Mu = num/den (each thread owns one (h,d) element), ||Mu_h||^2
  {
    const int m = tid >> 4, n = tid & 15;
    float s = 0.f;
#pragma unroll
    for (int w = 0; w < kWaves; ++w) s += sPart[w][m][n];
    const float den = sDen[m];
    sMu[m][n] = (den > 0.f) ? s / den : 0.f;
  }
  __syncthreads();
  if (tid < 16) {
    float s = 0.f;
#pragma unroll
    for (int d = 0; d < kD; ++d) { const float x = sMu[tid][d]; s += x * x; }
    sMuN2[tid] = s;
  }
  __syncthreads();

  // ---------- Phase 5: losses[h] = sum_i sqrt(||u_i||^2 - 2 u_i.Mu_h + ||Mu_h||^2)
  for (int tile = wave; tile < kTiles; tile += kWaves) {
    const int i0 = tile << 4;
    v8f c = {0.f, 0.f, 0.f, 0.f, 0.f, 0.f, 0.f, 0.f};
#pragma unroll
    for (int kk = 0; kk < 4; ++kk) {
      const int kb = (kk << 2) + hi2;       // this lane's two d indices
      v2f a, bm;
      a.x = sU[(i0 + l15) * kDStride + kb];     // A[M=i_local=l15][K=d]
      a.y = sU[(i0 + l15) * kDStride + kb + 1];
      bm.x = sMu[l15][kb];                      // B[K=d][N=h=l15]
      bm.y = sMu[l15][kb + 1];
      c = __builtin_amdgcn_wmma_f32_16x16x4_f32(false, a, false, bm,
                                                (short)0, c, false, false);
    }
    const int h = l15;                      // C[M=i_local][N=h]
    if (h < kH) {
      const float mun2 = sMuN2[h];
      float acc = 0.f;
#pragma unroll
      for (int v = 0; v < 8; ++v) {
        const int i = i0 + v + hi8;
        const float vn = sVn[i];
        const float d2 = vn * vn - 2.f * c[v] + mun2;
        acc += __builtin_sqrtf(fmaxf(d2, 0.f));  // zero-safe norm
      }
      atomicAdd(&sLoss[h], acc);
    }
  }
  __syncthreads();

  // ---------- Phase 6: argmin over h (first-min tiebreak, like jnp.argmin), emit Mu[:,h*]
  if (tid == 0) {
    int best = 0; float bl = sLoss[0];
    for (int h = 1; h < kH; ++h) { const float l = sLoss[h]; if (l < bl) { bl = l; best = h; } }
    sBest = best;
  }
  __syncthreads();
  if (tid < kD) out[(size_t)bo * kD + tid] = sMu[sBest][tid];
}

extern "C" void kernel_launch(void* const* d_in, const int* in_sizes, int n_in,
                              void* d_out, int out_size, void* d_ws, size_t ws_size,
                              hipStream_t stream) {
  (void)in_sizes; (void)n_in; (void)out_size; (void)d_ws; (void)ws_size;
  const float* u_predict  = (const float*)d_in[0];
  const int*   sample_idx = (const int*)d_in[1];   // JAX x64 disabled -> int32
  float*       out        = (float*)d_out;
  ransac_routing_kernel<<<kB * kO, kThreads, 0, stream>>>(u_predict, sample_idx, out);
}